// relative_depth_crit_14577119002949
// MI455X (gfx1250) — compile-verified
//
#include <hip/hip_runtime.h>

// Problem constants (from the reference): B=64, H=512, W=640, P=100000
#define PPB     100000
#define WIDTH   640
#define HW      (512 * 640)          // 327,680 elements per batch image
#define TOTAL   (64 * PPB)           // 6,400,000 points
#define NTHR    256
#define NBLK    3125
#define STRIDE  (NBLK * NTHR)        // 800,000 threads
#define ITERS   (TOTAL / STRIDE)     // exactly 8, no tail
#define MARGIN  1.0f

typedef float v2f __attribute__((ext_vector_type(2)));
typedef float v8f __attribute__((ext_vector_type(8)));

// Deterministic full-wave (32 lane) sum using V_WMMA_F32_16X16X4_F32.
// A-matrix (16x4 f32, 2 VGPRs): VGPR0 = per-lane value, VGPR1 = 0
//   -> A[m][0] = acc(lane m), A[m][2] = acc(lane m+16), other cols 0.
// B = all ones (4x16) -> D[m][n] = acc_m + acc_{m+16}  (= r_m, every column).
// Per-lane sum of the 8 D registers yields sum(r_0..7) on lanes 0-15 and
// sum(r_8..15) on lanes 16-31; a second identical WMMA adds those two
// halves, broadcasting the full 32-lane total into every D element.
// EXEC must be all ones at the call site (guaranteed: callers are fully
// converged). Hardware accumulation order is fixed -> deterministic.
__device__ __forceinline__ float wave_sum_wmma(float acc) {
  v2f a;    a[0] = acc;  a[1] = 0.0f;
  v2f ones; ones[0] = 1.0f; ones[1] = 1.0f;
  v8f c = {};
  v8f d = __builtin_amdgcn_wmma_f32_16x16x4_f32(false, a, false, ones,
                                                (short)0, c, false, false);
  float s = ((d[0] + d[1]) + (d[2] + d[3])) + ((d[4] + d[5]) + (d[6] + d[7]));
  v2f a2; a2[0] = s; a2[1] = 0.0f;
  v8f d2 = __builtin_amdgcn_wmma_f32_16x16x4_f32(false, a2, false, ones,
                                                 (short)0, c, false, false);
  return d2[0];
}

__global__ __launch_bounds__(NTHR)
void rdc_partial_kernel(const float* __restrict__ depth,
                        const int*   __restrict__ xA,
                        const int*   __restrict__ yA,
                        const int*   __restrict__ xB,
                        const int*   __restrict__ yB,
                        const int*   __restrict__ ord,
                        float*       __restrict__ partials) {
  const int tid = blockIdx.x * NTHR + threadIdx.x;   // 0 .. 799,999
  const int b0  = tid / PPB;                         // one umulhi, hoisted
  float acc = 0.0f;

#pragma unroll
  for (int k = 0; k < ITERS; ++k) {
    const int i = tid + k * STRIDE;                  // i = 800000*k + tid
    // batch index: i/PPB == 8*k + tid/PPB exactly (800000 = 8*100000)
    const int base = (8 * k + b0) * HW;

    if (k + 1 < ITERS) {                             // compile-time guard
      const int j = i + STRIDE;
      __builtin_prefetch(&xA[j], 0, 0);              // global_prefetch_b8
      __builtin_prefetch(&yA[j], 0, 0);
      __builtin_prefetch(&xB[j], 0, 0);
      __builtin_prefetch(&yB[j], 0, 0);
      __builtin_prefetch(&ord[j], 0, 0);
    }

    const int xa = xA[i], ya = yA[i];
    const int xb = xB[i], yb = yB[i];
    const int o  = ord[i];

    const float zA = depth[base + ya * WIDTH + xa];  // L2-resident gather
    const float zB = depth[base + yb * WIDTH + xb];

    const float zd   = zA - zB;
    const float gt   = (float)(o - 1);               // -1, 0, +1
    const float mask = fabsf(gt);                    //  1, 0,  1

    const float t   = fminf(gt * zd, MARGIN);
    const float lt  = __logf(1.0f + __expf(-t));     // logistic term
    const float sq  = fmaxf(zd * zd, MARGIN * MARGIN);

    acc += mask * lt + (1.0f - mask) * sq;           // branchless select
  }

  // ---- deterministic block reduction (8 waves of 32) ----
  const float wtot = wave_sum_wmma(acc);             // converged, EXEC full
  __shared__ float wsum[NTHR / 32];
  const int wave = threadIdx.x >> 5;
  if ((threadIdx.x & 31) == 0) wsum[wave] = wtot;
  __syncthreads();
  if (threadIdx.x == 0) {
    float s = 0.0f;
#pragma unroll
    for (int w = 0; w < NTHR / 32; ++w) s += wsum[w];
    partials[blockIdx.x] = s;
  }
}

__global__ __launch_bounds__(32)
void rdc_final_kernel(const float* __restrict__ partials,
                      float*       __restrict__ out) {
  float acc = 0.0f;
  for (int j = threadIdx.x; j < NBLK; j += 32) acc += partials[j];
  // all 32 lanes reconverge here; single wave -> EXEC all ones
  const float tot = wave_sum_wmma(acc);
  if (threadIdx.x == 0) out[0] = tot * (1.0f / (float)TOTAL);
}

extern "C" void kernel_launch(void* const* d_in, const int* in_sizes, int n_in,
                              void* d_out, int out_size, void* d_ws, size_t ws_size,
                              hipStream_t stream) {
  const float* depth = (const float*)d_in[0];   // (64,1,512,640) fp32
  const int*   xA    = (const int*)d_in[1];     // (64,100000) i32
  const int*   yA    = (const int*)d_in[2];
  const int*   xB    = (const int*)d_in[3];
  const int*   yB    = (const int*)d_in[4];
  const int*   ord   = (const int*)d_in[5];
  float*       out   = (float*)d_out;           // scalar
  float*       part  = (float*)d_ws;            // NBLK floats of scratch

  rdc_partial_kernel<<<NBLK, NTHR, 0, stream>>>(depth, xA, yA, xB, yB, ord, part);
  rdc_final_kernel<<<1, 32, 0, stream>>>(part, out);
}